// ResidualGatedGCN_18236431139071
// MI455X (gfx1250) — compile-verified
//
#include <hip/hip_runtime.h>
#include <hip/hip_bf16.h>

typedef __attribute__((ext_vector_type(2))) float v2f;
typedef __attribute__((ext_vector_type(8))) float v8f;

#define N_NODES_C 100000
#define N_EDGES_C 640000
#define DIM_C 128
#define PROJ_C 512

// ---------------------------------------------------------------------------
// Phase 1: proj = X @ W + b  (fp32 WMMA 16x16x4)
//   block = 256 threads = 8 waves; block covers rows [16*bid, 16*bid+16)
//   wave w covers columns [64w, 64w+64) -> 4 accumulators of 16x16
//   cols   0..127 -> h  (written to d_out)
//   cols 128..511 -> Q,K,V (written to workspace, each N*128 floats)
// ---------------------------------------------------------------------------
__global__ __launch_bounds__(256)
void gcn_gemm_wmma_f32(const float* __restrict__ X,
                       const float* __restrict__ W,
                       const float* __restrict__ bias,
                       float* __restrict__ outH,
                       float* __restrict__ qkv)
{
    const int lane    = threadIdx.x & 31;
    const int wave    = threadIdx.x >> 5;
    const int rowBase = blockIdx.x * 16;
    const int colBase = wave * 64;

    const int mn    = lane & 15;       // M index for A, N index for B/C/D
    const int kHalf = lane >> 4;       // selects K pair {0,1} vs {2,3}

    v8f acc0 = {}, acc1 = {}, acc2 = {}, acc3 = {};

    const float* __restrict__ Arow = X + (size_t)(rowBase + mn) * DIM_C;

    for (int k0 = 0; k0 < DIM_C; k0 += 4) {
        const int k = k0 + kHalf * 2;
        // A fragment: lane holds X[m][k], X[m][k+1] (contiguous -> b64 load)
        v2f a;
        a.x = Arow[k + 0];
        a.y = Arow[k + 1];

        const float* __restrict__ Wr0 = W + (size_t)(k + 0) * PROJ_C + colBase + mn;
        const float* __restrict__ Wr1 = W + (size_t)(k + 1) * PROJ_C + colBase + mn;

        v2f b0; b0.x = Wr0[ 0]; b0.y = Wr1[ 0];
        v2f b1; b1.x = Wr0[16]; b1.y = Wr1[16];
        v2f b2; b2.x = Wr0[32]; b2.y = Wr1[32];
        v2f b3; b3.x = Wr0[48]; b3.y = Wr1[48];

        acc0 = __builtin_amdgcn_wmma_f32_16x16x4_f32(false, a, false, b0,
                                                     (short)0, acc0, false, false);
        acc1 = __builtin_amdgcn_wmma_f32_16x16x4_f32(false, a, false, b1,
                                                     (short)0, acc1, false, false);
        acc2 = __builtin_amdgcn_wmma_f32_16x16x4_f32(false, a, false, b2,
                                                     (short)0, acc2, false, false);
        acc3 = __builtin_amdgcn_wmma_f32_16x16x4_f32(false, a, false, b3,
                                                     (short)0, acc3, false, false);
    }

    // Epilogue: add bias, route to h (d_out) or Q/K/V (workspace).
    // C/D layout: vgpr v, lane L -> M = v + (L>=16 ? 8 : 0), N = L%16.
    const int mOff = kHalf * 8;
    v8f accs[4] = {acc0, acc1, acc2, acc3};
#pragma unroll
    for (int nt = 0; nt < 4; ++nt) {
        const int col = colBase + nt * 16 + mn;
        const float bval = bias[col];
        float* __restrict__ dst =
            (col < DIM_C) ? outH
                          : qkv + (size_t)((col >> 7) - 1) * (size_t)N_NODES_C * DIM_C;
        const int c = col & (DIM_C - 1);
#pragma unroll
        for (int v = 0; v < 8; ++v) {
            const int row = rowBase + mOff + v;
            dst[(size_t)row * DIM_C + c] = accs[nt][v] + bval;
        }
    }
}

// ---------------------------------------------------------------------------
// Phase 2: per-edge gating + scatter-sum.
//   one edge per wave32; lane handles 4 contiguous dims (float4, coalesced).
//   eta = sigmoid(Q[r] + K[s]); atomicAdd(out[r], eta * V[s]).
// ---------------------------------------------------------------------------
__global__ __launch_bounds__(256)
void gcn_edge_scatter(const float* __restrict__ qkv,
                      const int* __restrict__ senders,
                      const int* __restrict__ receivers,
                      float* __restrict__ out)
{
    const int edge = blockIdx.x * 8 + (threadIdx.x >> 5);
    if (edge >= N_EDGES_C) return;
    const int lane = threadIdx.x & 31;

    const int s = senders[edge];
    const int r = receivers[edge];

    const float* __restrict__ Q = qkv;
    const float* __restrict__ K = qkv + (size_t)N_NODES_C * DIM_C;
    const float* __restrict__ V = qkv + 2 * (size_t)N_NODES_C * DIM_C;

    const int d = lane * 4;
    const float4 q  = *(const float4*)(Q + (size_t)r * DIM_C + d);
    const float4 kk = *(const float4*)(K + (size_t)s * DIM_C + d);
    const float4 vv = *(const float4*)(V + (size_t)s * DIM_C + d);

    float4 g;
    g.x = vv.x / (1.0f + expf(-(q.x + kk.x)));
    g.y = vv.y / (1.0f + expf(-(q.y + kk.y)));
    g.z = vv.z / (1.0f + expf(-(q.z + kk.z)));
    g.w = vv.w / (1.0f + expf(-(q.w + kk.w)));

    float* __restrict__ o = out + (size_t)r * DIM_C + d;
    atomicAdd(o + 0, g.x);
    atomicAdd(o + 1, g.y);
    atomicAdd(o + 2, g.z);
    atomicAdd(o + 3, g.w);
}

extern "C" void kernel_launch(void* const* d_in, const int* in_sizes, int n_in,
                              void* d_out, int out_size, void* d_ws, size_t ws_size,
                              hipStream_t stream) {
    const float* X        = (const float*)d_in[0];   // [N, 128]
    const float* Wk       = (const float*)d_in[1];   // [128, 512]
    const float* Wb       = (const float*)d_in[2];   // [512]
    const int* senders    = (const int*)d_in[3];     // [E]
    const int* receivers  = (const int*)d_in[4];     // [E]
    float* out            = (float*)d_out;           // [N, 128]
    float* qkv            = (float*)d_ws;            // 3 * N * 128 floats

    // Phase 1: full overwrite of d_out with h, Q/K/V into workspace.
    gcn_gemm_wmma_f32<<<N_NODES_C / 16, 256, 0, stream>>>(X, Wk, Wb, out, qkv);

    // Phase 2: gated scatter-sum into d_out (stream order guarantees phase 1 done).
    gcn_edge_scatter<<<(N_EDGES_C + 7) / 8, 256, 0, stream>>>(qkv, senders, receivers, out);
}